// FarthestPointSample_13434657702246
// MI455X (gfx1250) — compile-verified
//
#include <hip/hip_runtime.h>
#include <cstdint>

#define BATCHES   16
#define NPTS      65536
#define NCH       64
#define NTHREADS  1024
#define NWAVES    32          // 1024 / wave32
#define PPT       64          // points per thread (dist registers)
#define PPW       2048        // points per wave per step
#define CHUNK     64          // points per async chunk per wave
#define NCHUNK    32          // chunks per wave per step
#define CHUNK_B   (CHUNK*12)  // 768 bytes
#define MAX_S     3072

// Issue one 768B chunk of p (64 points * 12B) from global to this wave's LDS
// staging buffer via three async b64 transfers (32 lanes * 8B = 256B each).
__device__ __forceinline__ void async_issue_chunk(uint64_t gbase, uint32_t lds_base,
                                                  int w, int c, int buf, int lane)
{
    uint32_t voff0 = (uint32_t)((w * PPW + c * CHUNK) * 12) + (uint32_t)(lane * 8);
    uint32_t lds0  = lds_base + (uint32_t)((w * 2 + buf) * CHUNK_B) + (uint32_t)(lane * 8);
#pragma unroll
    for (int k = 0; k < 3; ++k) {
        uint32_t voff = voff0 + (uint32_t)(k * 256);
        uint32_t ldsa = lds0  + (uint32_t)(k * 256);
        asm volatile("global_load_async_to_lds_b64 %0, %1, %2"
                     :: "v"(ldsa), "v"(voff), "s"(gbase)
                     : "memory");
    }
}

__global__ __launch_bounds__(NTHREADS, 1)
void FarthestPointSample_kernel(const float* __restrict__ p,
                                const float* __restrict__ x,
                                const int*   __restrict__ nsamp,
                                float*       __restrict__ out)
{
    __shared__ float stage[NWAVES * 2 * CHUNK * 3];   // 48 KB double-buffered staging
    __shared__ int   idx_lds[MAX_S];                  // 12 KB chosen indices
    __shared__ float redv[NWAVES];
    __shared__ int   redi[NWAVES];

    const int tid  = threadIdx.x;
    const int lane = tid & 31;
    const int w    = tid >> 5;
    const int b    = blockIdx.x;

    int S = *nsamp;
    if (S > MAX_S) S = MAX_S;
    if (S < 1)     S = 1;

    const float* pb = p + (size_t)b * NPTS * 3;
    const uint64_t gbase      = (uint64_t)(uintptr_t)pb;
    const uint32_t stage_base = (uint32_t)(uintptr_t)(&stage[0]); // low 32 bits = LDS offset

    // Running min squared distance, resident in VGPRs.
    // Point index for register r: gi = w*2048 + r*32 + lane  (monotonic in r)
    float dist[PPT];
#pragma unroll
    for (int i = 0; i < PPT; ++i) dist[i] = __builtin_inff();

    if (tid == 0) idx_lds[0] = 0;          // reference always emits index 0 first

    int winner = 0;                        // index chosen in the previous step

    for (int s = 1; s < S; ++s) {
        // Kick chunk 0 first so the uniform coord fetch below overlaps its landing.
        async_issue_chunk(gbase, stage_base, w, 0, 0, lane);

        const int uw = __builtin_amdgcn_readfirstlane(winner);
        const float lx = pb[(size_t)uw * 3 + 0];
        const float ly = pb[(size_t)uw * 3 + 1];
        const float lz = pb[(size_t)uw * 3 + 2];

        float bestv = -__builtin_inff();

#pragma unroll
        for (int c = 0; c < NCHUNK; ++c) {
            const int buf = c & 1;
            if (c + 1 < NCHUNK) {
                async_issue_chunk(gbase, stage_base, w, c + 1, (c + 1) & 1, lane);
                asm volatile("s_wait_asynccnt 0x3" ::: "memory"); // chunk c landed (in-order)
            } else {
                asm volatile("s_wait_asynccnt 0x0" ::: "memory");
            }
#pragma unroll
            for (int j = 0; j < 2; ++j) {
                const int r = c * 2 + j;
                const float* sp = &stage[(w * 2 + buf) * (CHUNK * 3) + (j * 32 + lane) * 3];
                const float px = sp[0], py = sp[1], pz = sp[2];
                const float dx = px - lx, dy = py - ly, dz = pz - lz;
                const float d  = dx * dx + dy * dy + dz * dz;
                const float nd = fminf(dist[r], d);
                dist[r] = nd;
                bestv   = fmaxf(bestv, nd);     // value-only tracking in the hot loop
            }
        }

        // Register-only scan: smallest point index whose dist equals the max value.
        int besti = 0x7fffffff;
        const int gbaseidx = w * PPW + lane;
#pragma unroll
        for (int r = 0; r < PPT; ++r) {
            const int gi = gbaseidx + r * 32;
            if (dist[r] == bestv) besti = (gi < besti) ? gi : besti;
        }

        // Wave32 butterfly argmax (value desc, index asc => matches jnp.argmax tie-break)
#pragma unroll
        for (int off = 16; off > 0; off >>= 1) {
            const float ov = __shfl_xor(bestv, off, 32);
            const int   oi = __shfl_xor(besti, off, 32);
            if (ov > bestv || (ov == bestv && oi < besti)) { bestv = ov; besti = oi; }
        }

        __syncthreads();   // previous step's redv/redi reads are complete everywhere
        if (lane == 0) { redv[w] = bestv; redi[w] = besti; }
        __syncthreads();   // candidate table visible

        // Every wave redundantly reduces the 32 wave candidates -> identical winner.
        float v2 = redv[lane];
        int   i2 = redi[lane];
#pragma unroll
        for (int off = 16; off > 0; off >>= 1) {
            const float ov = __shfl_xor(v2, off, 32);
            const int   oi = __shfl_xor(i2, off, 32);
            if (ov > v2 || (ov == v2 && oi < i2)) { v2 = ov; i2 = oi; }
        }
        winner = i2;
        if (tid == 0) idx_lds[s] = winner;
    }

    __syncthreads();       // idx_lds fully written

    // ---- Gathers: p_s [B,S,3] then x_s [B,C,S], concatenated flat in d_out ----
    // Division-free loops: outer bounds are compile-time, inner strided by tid.
    float* out_p = out + (size_t)b * S * 3;                 // [S,3] slice
    float* out_x = out + (size_t)BATCHES * S * 3            // past p_s block
                       + (size_t)b * NCH * S;               // [C,S] slice
    const float* xb = x + (size_t)b * NCH * NPTS;

    for (int si = tid; si < S; si += NTHREADS) {
        const int gi = idx_lds[si];
        const float* src = pb + (size_t)gi * 3;
        float* dst = out_p + (size_t)si * 3;
        dst[0] = src[0];
        dst[1] = src[1];
        dst[2] = src[2];
    }

#pragma unroll 1
    for (int c = 0; c < NCH; ++c) {
        const float* xc = xb + (size_t)c * NPTS;
        float* oc = out_x + (size_t)c * S;
        for (int si = tid; si < S; si += NTHREADS) {
            oc[si] = xc[idx_lds[si]];
        }
    }
}

extern "C" void kernel_launch(void* const* d_in, const int* in_sizes, int n_in,
                              void* d_out, int out_size, void* d_ws, size_t ws_size,
                              hipStream_t stream)
{
    const float* p  = (const float*)d_in[0];   // [16, 65536, 3] f32
    const float* x  = (const float*)d_in[1];   // [16, 64, 65536] f32
    const int*   ns = (const int*)d_in[2];     // scalar num_samples (2048)
    (void)in_sizes; (void)n_in; (void)out_size; (void)d_ws; (void)ws_size;

    FarthestPointSample_kernel<<<BATCHES, NTHREADS, 0, stream>>>(p, x, ns, (float*)d_out);
}